// DADMMLRDiff_17368847745613
// MI455X (gfx1250) — compile-verified
//
#include <hip/hip_runtime.h>
#include <cstddef>

#define Pn   100
#define Bn   128
#define Nn   784
#define Kn   10
#define DEGf 4.0f
#define NV4  196              // Nn / 4 float4 chunks per row

#define AS1 __attribute__((address_space(1)))
#define AS3 __attribute__((address_space(3)))

typedef int v4i __attribute__((vector_size(16)));

#if defined(__AMDGCN__) && __has_builtin(__builtin_amdgcn_global_load_async_to_lds_b128)
#define USE_ASYNC 1
#else
#define USE_ASYNC 0
#endif

// One block per (color-node, batch sample). Phase 1: dot(x, a) over N with
// wave+LDS reduction, overlapped with async global->LDS staging of the 4
// neighbor rows and the mu row (ASYNCcnt path). Phase 2: elementwise primal
// update reading the staged rows from LDS.
__global__ __launch_bounds__(256)
void primal_color_kernel(
    const float* __restrict__ x, const float* __restrict__ y,
    const float* __restrict__ hk,                 // [P,6] this step's hyp
    const float* __restrict__ a_cur,              // old a (node p, same-parity nbrs)
    const float* __restrict__ a_cross,            // cross-parity nbr source
    float* __restrict__ a_nxt,
    const float* __restrict__ w_cur, const float* __restrict__ w_cross,
    float* __restrict__ w_nxt,
    const float* __restrict__ mu, const float* __restrict__ lam,
    const int* __restrict__ color_nodes,          // [P/2] node ids for this color
    const int* __restrict__ neighbors)            // [P,4]
{
  const int t  = threadIdx.x;
  const int b  = blockIdx.y;
  const int p  = color_nodes[blockIdx.x];
  const int pb = p * Bn + b;
  const size_t row = (size_t)pb * Nn;

  const int nb0 = neighbors[p*4+0], nb1 = neighbors[p*4+1];
  const int nb2 = neighbors[p*4+2], nb3 = neighbors[p*4+3];

  const float4* xr  = (const float4*)(x + row);
  const float4* ar  = (const float4*)(a_cur + row);
  const float4* mur = (const float4*)(mu + row);

  // parity select: +-2 neighbors share p's parity -> old buffer; +-1 -> cross
  const float4* n0 = (const float4*)(((((nb0 ^ p) & 1) == 0) ? a_cur : a_cross) + (size_t)(nb0*Bn + b) * Nn);
  const float4* n1 = (const float4*)(((((nb1 ^ p) & 1) == 0) ? a_cur : a_cross) + (size_t)(nb1*Bn + b) * Nn);
  const float4* n2 = (const float4*)(((((nb2 ^ p) & 1) == 0) ? a_cur : a_cross) + (size_t)(nb2*Bn + b) * Nn);
  const float4* n3 = (const float4*)(((((nb3 ^ p) & 1) == 0) ? a_cur : a_cross) + (size_t)(nb3*Bn + b) * Nn);

  __shared__ float s_red[8];
  __shared__ float s_xa;
#if USE_ASYNC
  __shared__ float4 s_nb[4][NV4];
  __shared__ float4 s_mu[NV4];
  if (t < NV4) {
    __builtin_amdgcn_global_load_async_to_lds_b128((AS1 v4i*)(n0 + t),  (AS3 v4i*)(&s_nb[0][t]), 0, 0);
    __builtin_amdgcn_global_load_async_to_lds_b128((AS1 v4i*)(n1 + t),  (AS3 v4i*)(&s_nb[1][t]), 0, 0);
    __builtin_amdgcn_global_load_async_to_lds_b128((AS1 v4i*)(n2 + t),  (AS3 v4i*)(&s_nb[2][t]), 0, 0);
    __builtin_amdgcn_global_load_async_to_lds_b128((AS1 v4i*)(n3 + t),  (AS3 v4i*)(&s_nb[3][t]), 0, 0);
    __builtin_amdgcn_global_load_async_to_lds_b128((AS1 v4i*)(mur + t), (AS3 v4i*)(&s_mu[t]),    0, 0);
  }
#endif

  // ---- phase 1: xa = dot(x_row, a_row) ----
  float4 xv = {0.f,0.f,0.f,0.f}, av = {0.f,0.f,0.f,0.f};
  float acc = 0.0f;
  if (t < NV4) {
    xv = xr[t]; av = ar[t];
    acc = xv.x*av.x + xv.y*av.y + xv.z*av.z + xv.w*av.w;
  }
  #pragma unroll
  for (int o = 16; o > 0; o >>= 1) acc += __shfl_down(acc, o);
  if ((t & 31) == 0) s_red[t >> 5] = acc;
  __syncthreads();
  if (t == 0) {
    float s = 0.0f;
    #pragma unroll
    for (int w = 0; w < 8; ++w) s += s_red[w];
    s_xa = s;
  }
  __syncthreads();
  const float xa = s_xa;

  const float wp = w_cur[pb];
  const float yv = y[pb];
  const float lv = lam[pb];
  const float h0 = fabsf(hk[p*6+0]);
  const float h1 = fabsf(hk[p*6+1]);
  const float h2 = fabsf(hk[p*6+2]);
  const float h5 = fabsf(hk[p*6+5]);
  const float coef = xa + wp - yv;

#if USE_ASYNC
#if __has_builtin(__builtin_amdgcn_s_wait_asynccnt)
  __builtin_amdgcn_s_wait_asynccnt(0);
#else
  asm volatile("s_wait_asynccnt 0" ::: "memory");
#endif
#endif

  // ---- phase 2: a_new = a - h1*(x*coef + h0*(4a - sum_nb) + 4*mu) ----
  if (t < NV4) {
    float4 s0, s1, s2, s3, m4;
#if USE_ASYNC
    s0 = s_nb[0][t]; s1 = s_nb[1][t]; s2 = s_nb[2][t]; s3 = s_nb[3][t]; m4 = s_mu[t];
#else
    s0 = n0[t]; s1 = n1[t]; s2 = n2[t]; s3 = n3[t]; m4 = mur[t];
#endif
    float4 out;
    out.x = av.x - h1*(xv.x*coef + h0*(DEGf*av.x - (s0.x+s1.x+s2.x+s3.x)) + DEGf*m4.x);
    out.y = av.y - h1*(xv.y*coef + h0*(DEGf*av.y - (s0.y+s1.y+s2.y+s3.y)) + DEGf*m4.y);
    out.z = av.z - h1*(xv.z*coef + h0*(DEGf*av.z - (s0.z+s1.z+s2.z+s3.z)) + DEGf*m4.z);
    out.w = av.w - h1*(xv.w*coef + h0*(DEGf*av.w - (s0.w+s1.w+s2.w+s3.w)) + DEGf*m4.w);
    ((float4*)(a_nxt + row))[t] = out;
  }

  if (t == 0) {
    float sw = 0.0f;
    sw += (((((nb0 ^ p) & 1) == 0) ? w_cur : w_cross)[nb0*Bn + b]);
    sw += (((((nb1 ^ p) & 1) == 0) ? w_cur : w_cross)[nb1*Bn + b]);
    sw += (((((nb2 ^ p) & 1) == 0) ? w_cur : w_cross)[nb2*Bn + b]);
    sw += (((((nb3 ^ p) & 1) == 0) ? w_cur : w_cross)[nb3*Bn + b]);
    const float gw = coef + h2*(DEGf*wp - sw) + DEGf*lv;
    w_nxt[pb] = wp - h5*gw;
  }
}

// Dual update: mu += |h3|*(4a - sum_nb a); lam += |h4|*(4w - sum_nb w)
__global__ __launch_bounds__(256)
void dual_kernel(const float* __restrict__ a_new, const float* __restrict__ w_new,
                 float* __restrict__ mu, float* __restrict__ lam,
                 const float* __restrict__ hk, const int* __restrict__ neighbors)
{
  const int t  = threadIdx.x;
  const int b  = blockIdx.y;
  const int p  = blockIdx.x;
  const int pb = p * Bn + b;
  const size_t row = (size_t)pb * Nn;

  const int nb0 = neighbors[p*4+0], nb1 = neighbors[p*4+1];
  const int nb2 = neighbors[p*4+2], nb3 = neighbors[p*4+3];

  const float4* ar = (const float4*)(a_new + row);
  const float4* n0 = (const float4*)(a_new + (size_t)(nb0*Bn + b) * Nn);
  const float4* n1 = (const float4*)(a_new + (size_t)(nb1*Bn + b) * Nn);
  const float4* n2 = (const float4*)(a_new + (size_t)(nb2*Bn + b) * Nn);
  const float4* n3 = (const float4*)(a_new + (size_t)(nb3*Bn + b) * Nn);
  float4* mur = (float4*)(mu + row);

  const float h3 = fabsf(hk[p*6+3]);

  if (t < NV4) {
    float4 av = ar[t], s0 = n0[t], s1 = n1[t], s2 = n2[t], s3 = n3[t];
    float4 m = mur[t];
    m.x += h3*(DEGf*av.x - (s0.x+s1.x+s2.x+s3.x));
    m.y += h3*(DEGf*av.y - (s0.y+s1.y+s2.y+s3.y));
    m.z += h3*(DEGf*av.z - (s0.z+s1.z+s2.z+s3.z));
    m.w += h3*(DEGf*av.w - (s0.w+s1.w+s2.w+s3.w));
    mur[t] = m;
  }
  if (t == 0) {
    const float h4 = fabsf(hk[p*6+4]);
    const float sw = w_new[nb0*Bn+b] + w_new[nb1*Bn+b] + w_new[nb2*Bn+b] + w_new[nb3*Bn+b];
    lam[pb] += h4*(DEGf*w_new[pb] - sw);
  }
}

__global__ __launch_bounds__(256)
void zero_f32(float* __restrict__ p, size_t n) {
  size_t i = (size_t)blockIdx.x * blockDim.x + threadIdx.x;
  size_t s = (size_t)gridDim.x * blockDim.x;
  for (; i < n; i += s) p[i] = 0.0f;
}

extern "C" void kernel_launch(void* const* d_in, const int* in_sizes, int n_in,
                              void* d_out, int out_size, void* d_ws, size_t ws_size,
                              hipStream_t stream) {
  (void)in_sizes; (void)n_in; (void)out_size; (void)ws_size;
  const float* x        = (const float*)d_in[0];  // inputs  [P,B,N]
  const float* y        = (const float*)d_in[1];  // labels  [P,B]
  const float* hyp      = (const float*)d_in[2];  // hyp     [K,P,6]
  const float* a0       = (const float*)d_in[3];  // a0      [P,B,N]
  const float* w0       = (const float*)d_in[4];  // omega0  [P,B]
  const int*   neighbors= (const int*)d_in[5];    // [P,4]
  const int*   colors   = (const int*)d_in[6];    // [2,P/2]

  const size_t nA = (size_t)Pn * Bn * Nn;         // 10,035,200
  const size_t nW = (size_t)Pn * Bn;              // 12,800

  float* outA = (float*)d_out;                    // final a lives here
  float* outW = outA + nA;                        // final omega lives here

  float* wsA  = (float*)d_ws;                     // a ping buffer   (40 MB)
  float* muB  = wsA + nA;                         // mu              (40 MB)
  float* wsW  = muB + nA;                         // omega ping
  float* lamB = wsW + nW;                         // lam

  // initialize state (deterministic per launch)
  (void)hipMemcpyAsync(outA, a0, nA * sizeof(float), hipMemcpyDeviceToDevice, stream);
  (void)hipMemcpyAsync(outW, w0, nW * sizeof(float), hipMemcpyDeviceToDevice, stream);
  zero_f32<<<2048, 256, 0, stream>>>(muB, nA);
  zero_f32<<<64,   256, 0, stream>>>(lamB, nW);

  const dim3 blk(256);
  const dim3 gridC(Pn / 2, Bn);
  const dim3 gridD(Pn, Bn);

  for (int k = 0; k < Kn; ++k) {
    const float* hk = hyp + (size_t)k * Pn * 6;
    float* aCur = (k & 1) ? wsA : outA;
    float* aNxt = (k & 1) ? outA : wsA;
    float* wCur = (k & 1) ? wsW : outW;
    float* wNxt = (k & 1) ? outW : wsW;

    // color 0 (even nodes): all reads from old state
    primal_color_kernel<<<gridC, blk, 0, stream>>>(
        x, y, hk, aCur, aCur, aNxt, wCur, wCur, wNxt, muB, lamB,
        colors, neighbors);
    // color 1 (odd nodes): cross-parity (+-1, even) reads freshly written state
    primal_color_kernel<<<gridC, blk, 0, stream>>>(
        x, y, hk, aCur, aNxt, aNxt, wCur, wNxt, wNxt, muB, lamB,
        colors + Pn / 2, neighbors);
    // dual update on fully-updated state
    dual_kernel<<<gridD, blk, 0, stream>>>(aNxt, wNxt, muB, lamB, hk, neighbors);
  }
  // K=10 is even: final state ended in outA/outW == d_out. No copy needed.
}